// FGHGNNConv_60404420051112
// MI455X (gfx1250) — compile-verified
//
#include <hip/hip_runtime.h>
#include <hip/hip_bf16.h>

typedef __attribute__((ext_vector_type(2))) float v2f;
typedef __attribute__((ext_vector_type(8))) float v8f;

#define AMODE_GINE  0   // Aeff = (1+eps)*A0 + A1
#define AMODE_MEAN  1   // Aeff = A0 / max(cnt,1)
#define AMODE_PLAIN 2   // Aeff = A0

// ---------------------------------------------------------------------------
// Scatter: GINE messages  agg[dst] += relu(x[src] + ea[e])
// one thread per (edge, 4-channel chunk)
// ---------------------------------------------------------------------------
__global__ __launch_bounds__(256)
void k_gine_scatter(const float* __restrict__ x, const float* __restrict__ ea,
                    const int* __restrict__ src, const int* __restrict__ dst,
                    float* __restrict__ agg, int E, int H, int chunkShift)
{
    long t = (long)blockIdx.x * 256 + threadIdx.x;
    long total = (long)E << chunkShift;
    if (t >= total) return;
    int e  = (int)(t >> chunkShift);
    int c4 = (((int)t) & ((1 << chunkShift) - 1)) << 2;
    int s = src[e], d = dst[e];
    float4 xv = *(const float4*)(x  + (long)s * H + c4);
    float4 ev = *(const float4*)(ea + (long)e * H + c4);
    float4 m;
    m.x = fmaxf(xv.x + ev.x, 0.0f);
    m.y = fmaxf(xv.y + ev.y, 0.0f);
    m.z = fmaxf(xv.z + ev.z, 0.0f);
    m.w = fmaxf(xv.w + ev.w, 0.0f);
    float* ap = agg + (long)d * H + c4;
    unsafeAtomicAdd(ap + 0, m.x);
    unsafeAtomicAdd(ap + 1, m.y);
    unsafeAtomicAdd(ap + 2, m.z);
    unsafeAtomicAdd(ap + 3, m.w);
}

// ---------------------------------------------------------------------------
// Scatter: SAGE  agg[dst] += x[src];  cnt[dst] += 1
// ---------------------------------------------------------------------------
__global__ __launch_bounds__(256)
void k_sage_scatter(const float* __restrict__ xsrc,
                    const int* __restrict__ src, const int* __restrict__ dst,
                    float* __restrict__ agg, float* __restrict__ cnt,
                    int E, int H, int chunkShift)
{
    long t = (long)blockIdx.x * 256 + threadIdx.x;
    long total = (long)E << chunkShift;
    if (t >= total) return;
    int e  = (int)(t >> chunkShift);
    int ch = (((int)t) & ((1 << chunkShift) - 1));
    int c4 = ch << 2;
    int s = src[e], d = dst[e];
    float4 xv = *(const float4*)(xsrc + (long)s * H + c4);
    float* ap = agg + (long)d * H + c4;
    unsafeAtomicAdd(ap + 0, xv.x);
    unsafeAtomicAdd(ap + 1, xv.y);
    unsafeAtomicAdd(ap + 2, xv.z);
    unsafeAtomicAdd(ap + 3, xv.w);
    if (ch == 0) unsafeAtomicAdd(cnt + d, 1.0f);
}

// ---------------------------------------------------------------------------
// Generic fp32 WMMA GEMM:  C[N x Pout] (+)= Aeff[N x K] @ W[K x Pout] + bias
// Aeff built on the fly from A0/A1/cnt per amode.  K, Pout powers of two,
// K <= 256, Pout multiple of 64.  Block: 256 thr = 8 waves -> 32x64 C tile.
// Uses V_WMMA_F32_16X16X4_F32 (exact fp32 on the matrix pipe).
// ---------------------------------------------------------------------------
__global__ __launch_bounds__(256)
void k_gemm(const float* __restrict__ A0, const float* __restrict__ A1,
            const float* __restrict__ epsPtr, const float* __restrict__ cnt,
            int amode,
            const float* __restrict__ W, const float* __restrict__ bias,
            float* __restrict__ C, int N, int K, int Pout, int accumulate)
{
    __shared__ float As[32 * 260];           // 32 rows x (K<=256 + 4 pad)
    const int ldsStride = K + 4;             // stride%64 == 4 -> conflict-free cols
    const int row0 = blockIdx.x * 32;
    const int col0 = blockIdx.y * 64;
    const int tid  = threadIdx.x;
    const int wave = tid >> 5;
    const int lane = tid & 31;

    float c0 = 1.0f;
    if (amode == AMODE_GINE) c0 = 1.0f + epsPtr[0];

    const int kshift = 31 - __builtin_clz(K);
    const int kmask  = K - 1;

    // cooperative load of the Aeff tile (zero-padded past N)
    for (int idx = tid; idx < 32 * K; idx += 256) {
        int r = idx >> kshift;
        int k = idx & kmask;
        int gr = row0 + r;
        float v = 0.0f;
        if (gr < N) {
            long o = (long)gr * K + k;
            if (amode == AMODE_GINE) {
                v = c0 * A0[o] + A1[o];
            } else if (amode == AMODE_MEAN) {
                float cc = cnt[gr];
                cc = cc > 1.0f ? cc : 1.0f;
                v = A0[o] / cc;
            } else {
                v = A0[o];
            }
        }
        As[r * ldsStride + k] = v;
    }
    __syncthreads();

    // wave's 16x16 tile within the 32x64 block tile
    const int wr = (wave >> 2) << 4;         // 0 or 16
    const int wc = (wave & 3) << 4;          // 0,16,32,48
    const int m  = lane & 15;                // A row / B col / C col
    const int kb = (lane >> 4) << 1;         // 0 or 2 (ISA half-wave K split)

    v8f acc = {};
    for (int k0 = 0; k0 < K; k0 += 4) {
        v2f a, b;
        const float* ap = &As[(wr + m) * ldsStride + k0 + kb];
        a.x = ap[0];
        a.y = ap[1];
        const float* bp = &W[(long)(k0 + kb) * Pout + col0 + wc + m];
        b.x = bp[0];
        b.y = bp[Pout];
        acc = __builtin_amdgcn_wmma_f32_16x16x4_f32(
            false, a, false, b, (short)0, acc, false, false);
    }

    const int colg = col0 + wc + m;
    const float bv = bias[colg];
    const int rbase = row0 + wr + ((lane >> 4) << 3);
#pragma unroll
    for (int v = 0; v < 8; v++) {
        int rowg = rbase + v;
        if (rowg < N) {
            long off = (long)rowg * Pout + colg;
            float val = acc[v] + bv;
            if (accumulate) val += C[off];
            C[off] = val;
        }
    }
}

// ---------------------------------------------------------------------------
// Per-column sum / sum-of-squares over N rows (P == blockDim.x == 256)
// ---------------------------------------------------------------------------
__global__ __launch_bounds__(256)
void k_col_stats(const float* __restrict__ h, int N, int P,
                 float* __restrict__ sum, float* __restrict__ sumsq)
{
    int col  = threadIdx.x;
    long r0  = (long)blockIdx.x * 256;
    float s = 0.0f, ss = 0.0f;
    for (int i = 0; i < 256; i++) {
        long r = r0 + i;
        if (r < N) {
            float v = h[r * P + col];
            s  += v;
            ss += v * v;
        }
    }
    unsafeAtomicAdd(&sum[col],   s);
    unsafeAtomicAdd(&sumsq[col], ss);
}

// ---------------------------------------------------------------------------
// BatchNorm (batch stats, biased var) + ReLU, in place
// ---------------------------------------------------------------------------
__global__ __launch_bounds__(256)
void k_bn_relu(float* __restrict__ h,
               const float* __restrict__ sum, const float* __restrict__ sumsq,
               const float* __restrict__ gamma, const float* __restrict__ beta,
               int N, int P)
{
    long t = (long)blockIdx.x * 256 + threadIdx.x;
    long total = (long)N * P;
    if (t >= total) return;
    int col = (int)(t & (P - 1));
    float invN = 1.0f / (float)N;
    float mu  = sum[col] * invN;
    float var = sumsq[col] * invN - mu * mu;
    float sc  = gamma[col] * rsqrtf(var + 1e-5f);
    float sh  = beta[col] - mu * sc;
    float v = h[t] * sc + sh;
    h[t] = v > 0.0f ? v : 0.0f;
}

// ---------------------------------------------------------------------------
extern "C" void kernel_launch(void* const* d_in, const int* in_sizes, int n_in,
                              void* d_out, int out_size, void* d_ws, size_t ws_size,
                              hipStream_t stream)
{
    const float* x      = (const float*)d_in[0];
    const float* ea     = (const float*)d_in[1];
    const float* xcl    = (const float*)d_in[2];
    const float* eacl   = (const float*)d_in[3];
    const int*   ei_a   = (const int*)d_in[4];
    const int*   ei_c   = (const int*)d_in[5];
    const int*   ei_a2c = (const int*)d_in[6];
    const int*   ei_c2a = (const int*)d_in[7];
    const float* atom_eps = (const float*)d_in[8];
    const float* atom_W   = (const float*)d_in[9];
    const float* atom_b   = (const float*)d_in[10];
    const float* atom_g   = (const float*)d_in[11];
    const float* atom_be  = (const float*)d_in[12];
    const float* cl_eps = (const float*)d_in[13];
    const float* cl_W   = (const float*)d_in[14];
    const float* cl_b   = (const float*)d_in[15];
    const float* cl_g   = (const float*)d_in[16];
    const float* cl_be  = (const float*)d_in[17];
    const float* a2c_Wl = (const float*)d_in[18];
    const float* a2c_bl = (const float*)d_in[19];
    const float* a2c_Wr = (const float*)d_in[20];
    const float* a2c_br = (const float*)d_in[21];
    const float* c2a_Wl = (const float*)d_in[22];
    const float* c2a_bl = (const float*)d_in[23];
    const float* c2a_Wr = (const float*)d_in[24];
    const float* c2a_br = (const float*)d_in[25];
    const float* mA_W = (const float*)d_in[26];
    const float* mA_b = (const float*)d_in[27];
    const float* mC_W = (const float*)d_in[28];
    const float* mC_b = (const float*)d_in[29];

    const int P  = in_sizes[10];          // 256
    const int H  = in_sizes[27];          // 128
    const int NA = in_sizes[0] / H;       // 100000
    const int NC = in_sizes[2] / H;       // 25000
    const int EA = in_sizes[4] / 2;       // 600000
    const int EC = in_sizes[5] / 2;       // 100000
    const int EB1 = in_sizes[6] / 2;      // 100000
    const int EB2 = in_sizes[7] / 2;      // 100000

    int chunkShift = 0;
    while ((1 << chunkShift) < (H >> 2)) chunkShift++;   // log2(H/4)

    // ---- workspace layout (floats); atomic targets first for one memset ----
    float* ws = (float*)d_ws;
    size_t o = 0;
    float* aggA  = ws + o; o += (size_t)NA * H;
    float* aggC  = ws + o; o += (size_t)NC * H;
    float* saggC = ws + o; o += (size_t)NC * H;   // atom -> cluster aggregate
    float* saggA = ws + o; o += (size_t)NA * H;   // cluster -> atom aggregate
    float* cntC  = ws + o; o += (size_t)NC;
    float* cntA  = ws + o; o += (size_t)NA;
    float* stats = ws + o; o += (size_t)4 * P;    // sumA, sqA, sumC, sqC
    size_t zeroFloats = o;
    float* hA = ws + o; o += (size_t)NA * P;
    float* hC = ws + o; o += (size_t)NC * P;

    float* outA = (float*)d_out;                  // [NA, H]
    float* outC = outA + (size_t)NA * H;          // [NC, H]

    hipMemsetAsync(ws, 0, zeroFloats * sizeof(float), stream);

    const int TB = 256;
    // ---- scatters ----
    {
        long tA = (long)EA << chunkShift;
        k_gine_scatter<<<(int)((tA + TB - 1) / TB), TB, 0, stream>>>(
            x, ea, ei_a, ei_a + EA, aggA, EA, H, chunkShift);
        long tC = (long)EC << chunkShift;
        k_gine_scatter<<<(int)((tC + TB - 1) / TB), TB, 0, stream>>>(
            xcl, eacl, ei_c, ei_c + EC, aggC, EC, H, chunkShift);
        long t1 = (long)EB1 << chunkShift;
        k_sage_scatter<<<(int)((t1 + TB - 1) / TB), TB, 0, stream>>>(
            x, ei_a2c, ei_a2c + EB1, saggC, cntC, EB1, H, chunkShift);
        long t2 = (long)EB2 << chunkShift;
        k_sage_scatter<<<(int)((t2 + TB - 1) / TB), TB, 0, stream>>>(
            xcl, ei_c2a, ei_c2a + EB2, saggA, cntA, EB2, H, chunkShift);
    }

    dim3 gA((NA + 31) / 32, P / 64);
    dim3 gC((NC + 31) / 32, P / 64);

    // ---- GINE linears (pre-BN) ----
    k_gemm<<<gA, TB, 0, stream>>>(x,   aggA, atom_eps, nullptr, AMODE_GINE,
                                  atom_W, atom_b, hA, NA, H, P, 0);
    k_gemm<<<gC, TB, 0, stream>>>(xcl, aggC, cl_eps,   nullptr, AMODE_GINE,
                                  cl_W, cl_b, hC, NC, H, P, 0);

    // ---- BN stats + apply (+ReLU) ----
    k_col_stats<<<(NA + 255) / 256, 256, 0, stream>>>(hA, NA, P, stats,         stats + P);
    k_col_stats<<<(NC + 255) / 256, 256, 0, stream>>>(hC, NC, P, stats + 2 * P, stats + 3 * P);
    {
        long tA = (long)NA * P, tC = (long)NC * P;
        k_bn_relu<<<(int)((tA + 255) / 256), 256, 0, stream>>>(
            hA, stats, stats + P, atom_g, atom_be, NA, P);
        k_bn_relu<<<(int)((tC + 255) / 256), 256, 0, stream>>>(
            hC, stats + 2 * P, stats + 3 * P, cl_g, cl_be, NC, P);
    }

    // ---- SAGE convs, accumulated onto post-BN features ----
    // atoms: hA += mean(saggA)@c2a_Wl + c2a_bl + x@c2a_Wr + c2a_br
    k_gemm<<<gA, TB, 0, stream>>>(saggA, nullptr, nullptr, cntA, AMODE_MEAN,
                                  c2a_Wl, c2a_bl, hA, NA, H, P, 1);
    k_gemm<<<gA, TB, 0, stream>>>(x, nullptr, nullptr, nullptr, AMODE_PLAIN,
                                  c2a_Wr, c2a_br, hA, NA, H, P, 1);
    // clusters: hC += mean(saggC)@a2c_Wl + a2c_bl + xcl@a2c_Wr + a2c_br
    k_gemm<<<gC, TB, 0, stream>>>(saggC, nullptr, nullptr, cntC, AMODE_MEAN,
                                  a2c_Wl, a2c_bl, hC, NC, H, P, 1);
    k_gemm<<<gC, TB, 0, stream>>>(xcl, nullptr, nullptr, nullptr, AMODE_PLAIN,
                                  a2c_Wr, a2c_br, hC, NC, H, P, 1);

    // ---- merge linears -> d_out ----
    dim3 mA((NA + 31) / 32, H / 64);
    dim3 mC((NC + 31) / 32, H / 64);
    k_gemm<<<mA, TB, 0, stream>>>(hA, nullptr, nullptr, nullptr, AMODE_PLAIN,
                                  mA_W, mA_b, outA, NA, P, H, 0);
    k_gemm<<<mC, TB, 0, stream>>>(hC, nullptr, nullptr, nullptr, AMODE_PLAIN,
                                  mC_W, mC_b, outC, NC, P, H, 0);
}